// myModel_77111842832559
// MI455X (gfx1250) — compile-verified
//
#include <hip/hip_runtime.h>

typedef __attribute__((ext_vector_type(2))) float v2f;
typedef __attribute__((ext_vector_type(8))) float v8f;

#define TPB 256
static inline unsigned cdiv(unsigned a, unsigned b) { return (a + b - 1) / b; }

// ---------------- elementwise / setup kernels ----------------

__global__ void fill_kernel(float* __restrict__ p, float v, int n) {
    int i = blockIdx.x * blockDim.x + threadIdx.x;
    if (i < n) p[i] = v;
}

// deg[dst] += 1 for every edge (deg pre-filled with 1.0 = self loop)
__global__ void deg_kernel(const int* __restrict__ dst, float* __restrict__ deg, int E) {
    int e = blockIdx.x * blockDim.x + threadIdx.x;
    if (e < E) atomicAdd(&deg[dst[e]], 1.0f);
}

__global__ void rsqrt_kernel(float* __restrict__ d, int n) {
    int i = blockIdx.x * blockDim.x + threadIdx.x;
    if (i < n) d[i] = rsqrtf(d[i]);   // deg >= 1 always (self loops)
}

__global__ void norm_kernel(const int* __restrict__ src, const int* __restrict__ dst,
                            const float* __restrict__ dinv, float* __restrict__ nrm, int E) {
    int e = blockIdx.x * blockDim.x + threadIdx.x;
    if (e < E) nrm[e] = dinv[src[e]] * dinv[dst[e]];
}

// layer 1 matmul: t[n, j] = x[n,0]*W1[0,j] + x[n,1]*W1[1,j]   (K=2, VALU is right)
__global__ void l1mm_kernel(const float* __restrict__ x, const float* __restrict__ W1,
                            float* __restrict__ t, int total) {
    int i = blockIdx.x * blockDim.x + threadIdx.x;
    if (i >= total) return;
    int n = i >> 6, j = i & 63;
    t[i] = fmaf(x[2 * n], W1[j], x[2 * n + 1] * W1[64 + j]);
}

// agg[n,j] = t[n,j] * dinv[n]^2 + b[j]   (self-loop contribution + bias as init)
__global__ void initself64_kernel(const float* __restrict__ t, const float* __restrict__ dinv,
                                  const float* __restrict__ b, float* __restrict__ agg, int total) {
    int i = blockIdx.x * blockDim.x + threadIdx.x;
    if (i >= total) return;
    int n = i >> 6, j = i & 63;
    float dv = dinv[n];
    agg[i] = fmaf(t[i], dv * dv, b[j]);
}

// edge scatter: 16 lanes per edge, float4 gather + 4 atomic f32 adds (L2-resident)
__global__ void scatter64_kernel(const int* __restrict__ src, const int* __restrict__ dst,
                                 const float* __restrict__ nrm, const float* __restrict__ t,
                                 float* __restrict__ agg, int E) {
    int tid = blockIdx.x * blockDim.x + threadIdx.x;
    int e = tid >> 4;
    if (e >= E) return;
    int q = (tid & 15) << 2;            // feature offset 0..60 step 4
    int s = src[e], d = dst[e];
    float w = nrm[e];
    float4 v = *reinterpret_cast<const float4*>(t + s * 64 + q);
    float* base = agg + d * 64 + q;
    atomicAdd(base + 0, v.x * w);
    atomicAdd(base + 1, v.y * w);
    atomicAdd(base + 2, v.z * w);
    atomicAdd(base + 3, v.w * w);
}

__global__ void relu_kernel(float* __restrict__ p, int n) {
    int i = blockIdx.x * blockDim.x + threadIdx.x;
    if (i < n) p[i] = fmaxf(p[i], 0.0f);
}

// ---------------- layer 2 GEMM: [N,64] x [64,64] via V_WMMA_F32_16X16X4_F32 ----------------
// One wave32 computes a 16x64 output tile: 4 accumulators (16x16 f32, v8f each),
// 16 k-steps of K=4  => 64 v_wmma_f32_16x16x4_f32 per wave. Exact fp32 math.
__global__ void gemm64_wmma_kernel(const float* __restrict__ H, const float* __restrict__ W,
                                   float* __restrict__ T, int nrows) {
    int wave = (blockIdx.x * blockDim.x + threadIdx.x) >> 5;
    int lane = threadIdx.x & 31;
    int row0 = wave * 16;
    if (row0 >= nrows) return;        // wave-uniform: EXEC stays all-ones for WMMA

    int m  = lane & 15;               // M (A) / N (B,C) index within tile
    int hk = lane >> 4;               // half-wave selects K pair / row group

    v8f acc0 = {}, acc1 = {}, acc2 = {}, acc3 = {};
    const float* hrow = H + (row0 + m) * 64;

#pragma unroll
    for (int k0 = 0; k0 < 64; k0 += 4) {
        // A 16x4 fp32: VGPR0 = K=k0+{0,2}, VGPR1 = K=k0+{1,3}, lanes = M
        v2f a;
        a.x = hrow[k0 + 2 * hk + 0];
        a.y = hrow[k0 + 2 * hk + 1];
        // B 4x16 fp32 per N-tile: VGPR j holds rows K=j and K=j+2, lanes = N
        v2f b0, b1, b2, b3;
        const float* w0 = W + (k0 + 2 * hk) * 64 + m;     // row k0+2hk,   col m (+tile)
        const float* w1 = W + (k0 + 2 * hk + 1) * 64 + m; // row k0+2hk+1
        b0.x = w0[0];  b0.y = w1[0];
        b1.x = w0[16]; b1.y = w1[16];
        b2.x = w0[32]; b2.y = w1[32];
        b3.x = w0[48]; b3.y = w1[48];
        acc0 = __builtin_amdgcn_wmma_f32_16x16x4_f32(false, a, false, b0, (short)0, acc0, false, false);
        acc1 = __builtin_amdgcn_wmma_f32_16x16x4_f32(false, a, false, b1, (short)0, acc1, false, false);
        acc2 = __builtin_amdgcn_wmma_f32_16x16x4_f32(false, a, false, b2, (short)0, acc2, false, false);
        acc3 = __builtin_amdgcn_wmma_f32_16x16x4_f32(false, a, false, b3, (short)0, acc3, false, false);
    }

    // C/D 16x16 f32 layout: VGPR j -> M = row0 + j + 8*hk, N = m
#pragma unroll
    for (int j = 0; j < 8; ++j) {
        float* orow = T + (row0 + j + 8 * hk) * 64 + m;
        orow[0]  = acc0[j];
        orow[16] = acc1[j];
        orow[32] = acc2[j];
        orow[48] = acc3[j];
    }
}

// layer 3: t3[n] = dot(h2[n,:], W3[:,0])
__global__ void l3_kernel(const float* __restrict__ h, const float* __restrict__ W3,
                          float* __restrict__ t3, int N) {
    int n = blockIdx.x * blockDim.x + threadIdx.x;
    if (n >= N) return;
    const float* row = h + n * 64;
    float acc = 0.0f;
#pragma unroll
    for (int k = 0; k < 64; ++k) acc = fmaf(row[k], W3[k], acc);
    t3[n] = acc;
}

__global__ void outinit_kernel(const float* __restrict__ t3, const float* __restrict__ dinv,
                               const float* __restrict__ b3, float* __restrict__ out, int N) {
    int n = blockIdx.x * blockDim.x + threadIdx.x;
    if (n >= N) return;
    float dv = dinv[n];
    out[n] = fmaf(t3[n], dv * dv, b3[0]);
}

__global__ void scatter1_kernel(const int* __restrict__ src, const int* __restrict__ dst,
                                const float* __restrict__ nrm, const float* __restrict__ t3,
                                float* __restrict__ out, int E) {
    int e = blockIdx.x * blockDim.x + threadIdx.x;
    if (e < E) atomicAdd(&out[dst[e]], t3[src[e]] * nrm[e]);
}

// ---------------- launch ----------------

extern "C" void kernel_launch(void* const* d_in, const int* in_sizes, int n_in,
                              void* d_out, int out_size, void* d_ws, size_t ws_size,
                              hipStream_t stream) {
    const float* x  = (const float*)d_in[0];
    const int*   ei = (const int*)d_in[1];
    const float* W1 = (const float*)d_in[2];
    const float* b1 = (const float*)d_in[3];
    const float* W2 = (const float*)d_in[4];
    const float* b2 = (const float*)d_in[5];
    const float* W3 = (const float*)d_in[6];
    const float* b3 = (const float*)d_in[7];
    float* out = (float*)d_out;

    const int N = in_sizes[0] / 2;     // x is [N,2]
    const int E = in_sizes[1] / 2;     // edge_index is [2,E]
    const int* src = ei;
    const int* dst = ei + E;

    // workspace layout (floats): bufA[N*64] | bufB[N*64] | dinv[N] | norm[E] | t3[N]
    float* bufA = (float*)d_ws;
    float* bufB = bufA + (size_t)N * 64;
    float* dinv = bufB + (size_t)N * 64;
    float* nrm  = dinv + N;
    float* t3   = nrm + E;

    const int NF = N * 64;

    // degrees -> dinv -> per-edge norm
    fill_kernel<<<cdiv(N, TPB), TPB, 0, stream>>>(dinv, 1.0f, N);          // self loop
    deg_kernel<<<cdiv(E, TPB), TPB, 0, stream>>>(dst, dinv, E);
    rsqrt_kernel<<<cdiv(N, TPB), TPB, 0, stream>>>(dinv, N);
    norm_kernel<<<cdiv(E, TPB), TPB, 0, stream>>>(src, dst, dinv, nrm, E);

    // ---- layer 1: t1 = x@W1 (bufA); h1 = relu(prop(t1)+b1) (bufB)
    l1mm_kernel<<<cdiv(NF, TPB), TPB, 0, stream>>>(x, W1, bufA, NF);
    initself64_kernel<<<cdiv(NF, TPB), TPB, 0, stream>>>(bufA, dinv, b1, bufB, NF);
    scatter64_kernel<<<cdiv(E * 16, TPB), TPB, 0, stream>>>(src, dst, nrm, bufA, bufB, E);
    relu_kernel<<<cdiv(NF, TPB), TPB, 0, stream>>>(bufB, NF);

    // ---- layer 2: t2 = h1@W2 via WMMA (bufA); h2 = relu(prop(t2)+b2) (bufB)
    {
        unsigned waves = (unsigned)(N + 15) / 16;               // N divisible by 16 here
        unsigned blocks = cdiv(waves * 32, TPB);
        gemm64_wmma_kernel<<<blocks, TPB, 0, stream>>>(bufB, W2, bufA, N);
    }
    initself64_kernel<<<cdiv(NF, TPB), TPB, 0, stream>>>(bufA, dinv, b2, bufB, NF);
    scatter64_kernel<<<cdiv(E * 16, TPB), TPB, 0, stream>>>(src, dst, nrm, bufA, bufB, E);
    relu_kernel<<<cdiv(NF, TPB), TPB, 0, stream>>>(bufB, NF);

    // ---- layer 3: t3 = h2@W3; out = prop(t3)+b3 (no relu)
    l3_kernel<<<cdiv(N, TPB), TPB, 0, stream>>>(bufB, W3, t3, N);
    outinit_kernel<<<cdiv(N, TPB), TPB, 0, stream>>>(t3, dinv, b3, out, N);
    scatter1_kernel<<<cdiv(E, TPB), TPB, 0, stream>>>(src, dst, nrm, t3, out, E);
}